// SpectralConvWavelet2DCwt_16853451669622
// MI455X (gfx1250) — compile-verified
//
#include <hip/hip_runtime.h>

typedef __attribute__((ext_vector_type(2))) float v2f;
typedef __attribute__((ext_vector_type(8))) float v8f;

namespace {

constexpr int kB = 32, kCi = 32, kCo = 32, kH = 64, kW = 64, kH2 = 32, kW2 = 32;
constexpr int kImg = kH * kW;                  // 4096
constexpr int kQuads = kH2 * kW2;              // 1024
constexpr int kHiStride = 6 * kQuads * 2;      // 12288 floats per (b,c) image of highs
constexpr int kLds = kW + 4;                   // 68: rows 16B-aligned, banks shift by 4/row
constexpr float kS = 0.70710678118654752440f;  // 1/sqrt(2)

__constant__ float cH0[13] = {
    -0.0017578125f, 0.0f, 0.022265625f, -0.046875f, -0.0482421875f, 0.296875f,
    0.55546875f, 0.296875f, -0.0482421875f, -0.046875f, 0.022265625f, 0.0f,
    -0.0017578125f};
__constant__ float cH1[19] = {
    -7.062639e-05f, 0.0f, 0.0013419f, -0.0018834f, -0.0071568f, 0.023856f,
    0.0556431f, -0.0516881f, -0.2997576f, 0.5594308f, -0.2997576f, -0.0516881f,
    0.0556431f, 0.023856f, -0.0071568f, -0.0018834f, 0.0013419f, 0.0f,
    -7.062639e-05f};
// G0 = (-1)^(n+1) * H1
__constant__ float cG0[19] = {
    7.062639e-05f, 0.0f, -0.0013419f, -0.0018834f, 0.0071568f, 0.023856f,
    -0.0556431f, -0.0516881f, 0.2997576f, 0.5594308f, 0.2997576f, -0.0516881f,
    -0.0556431f, 0.023856f, 0.0071568f, -0.0018834f, -0.0013419f, 0.0f,
    7.062639e-05f};
// G1 = (-1)^n * H0
__constant__ float cG1[13] = {
    -0.0017578125f, 0.0f, 0.022265625f, 0.046875f, -0.0482421875f, -0.296875f,
    0.55546875f, -0.296875f, -0.0482421875f, 0.046875f, 0.022265625f, 0.0f,
    -0.0017578125f};

__device__ __forceinline__ int symi(int i) {
  i = (i < 0) ? (-1 - i) : i;
  return (i >= 64) ? (127 - i) : i;
}

// Async global->LDS copy of 16 bytes (ASYNCcnt-tracked), per CDNA5 ISA.
__device__ __forceinline__ void async_load_b128(const void* gsrc, void* ldst) {
  unsigned laddr =
      (unsigned)(uintptr_t)(__attribute__((address_space(3))) const void*)ldst;
  unsigned long long gaddr = (unsigned long long)(uintptr_t)gsrc;
  asm volatile("global_load_async_to_lds_b128 %0, %1, off" ::"v"(laddr),
               "v"(gaddr)
               : "memory");
}
__device__ __forceinline__ void wait_asynccnt0() {
  asm volatile("s_wait_asynccnt 0" ::: "memory");
}

// ---------------------------------------------------------------------------
// Kernel 1: forward DTCWT (J=1). One workgroup per (b, ci) image.
// Writes ll[b,ci,64,64] and highs[b,ci,6,32,32,2] (orientation order
// d15,d45,d75,d105,d135,d165).
// ---------------------------------------------------------------------------
__global__ __launch_bounds__(256) void fwd_dtcwt(const float* __restrict__ x,
                                                 float* __restrict__ ll,
                                                 float* __restrict__ highs) {
  __shared__ float img[kH][kLds];
  __shared__ float lo[kH][kLds];
  __shared__ float hi[kH][kLds];
  const int image = blockIdx.x;  // b*kCi + ci
  const int t = threadIdx.x;
  const float* xin = x + (size_t)image * kImg;

  // Stage the 64x64 image into LDS with async b128 copies.
  for (int idx = t * 4; idx < kImg; idx += 1024) {
    async_load_b128(xin + idx, &img[idx >> 6][idx & 63]);
  }
  wait_asynccnt0();
  __syncthreads();

  // Row filters along W (symmetric padding).
  for (int idx = t; idx < kImg; idx += 256) {
    const int r = idx >> 6, c = idx & 63;
    float aLo = 0.f, aHi = 0.f;
#pragma unroll
    for (int k = 0; k < 13; ++k) aLo += cH0[k] * img[r][symi(c - 6 + k)];
#pragma unroll
    for (int k = 0; k < 19; ++k) aHi += cH1[k] * img[r][symi(c - 9 + k)];
    lo[r][c] = aLo;
    hi[r][c] = aHi;
  }
  __syncthreads();

  // ll = colf(lo, H0) at full resolution; vectorized b128 stores.
  float* llp = ll + (size_t)image * kImg;
  for (int idx = t * 4; idx < kImg; idx += 1024) {
    const int r = idx >> 6, c0 = idx & 63;
    float acc[4] = {0.f, 0.f, 0.f, 0.f};
#pragma unroll
    for (int k = 0; k < 13; ++k) {
      const int rr = symi(r - 6 + k);
      const float w = cH0[k];
#pragma unroll
      for (int j = 0; j < 4; ++j) acc[j] += w * lo[rr][c0 + j];
    }
    float4 v4 = {acc[0], acc[1], acc[2], acc[3]};
    *(float4*)(llp + idx) = v4;
  }

  // Highpass bands + q2c, one 2x2 quad per iteration; b64 (re,im) stores.
  float* hp = highs + (size_t)image * kHiStride;
  for (int q = t; q < kQuads; q += 256) {
    const int qi = q >> 5, qj = q & 31;
    float lhv[2][2], hlv[2][2], hhv[2][2];
#pragma unroll
    for (int dr = 0; dr < 2; ++dr) {
#pragma unroll
      for (int dc = 0; dc < 2; ++dc) {
        const int r = 2 * qi + dr, c = 2 * qj + dc;
        float a1 = 0.f, a0 = 0.f, a2 = 0.f;
#pragma unroll
        for (int k = 0; k < 19; ++k) {
          const int rr = symi(r - 9 + k);
          a1 += cH1[k] * lo[rr][c];  // lh = colf(lo, H1)
          a2 += cH1[k] * hi[rr][c];  // hh = colf(hi, H1)
        }
#pragma unroll
        for (int k = 0; k < 13; ++k) a0 += cH0[k] * hi[symi(r - 6 + k)][c];
        lhv[dr][dc] = a1;
        hhv[dr][dc] = a2;
        hlv[dr][dc] = a0;
      }
    }
    // q2c: (a,b,c,d) -> first=(a-d, b+c) second=(a+d, b-c), scaled by 1/sqrt2
    float a, b, c2, d;
    a = lhv[0][0] * kS; b = lhv[0][1] * kS; c2 = lhv[1][0] * kS; d = lhv[1][1] * kS;
    *(float2*)(hp + (0 * kQuads + q) * 2) = make_float2(a - d, b + c2);  // d15
    *(float2*)(hp + (5 * kQuads + q) * 2) = make_float2(a + d, b - c2);  // d165
    a = hhv[0][0] * kS; b = hhv[0][1] * kS; c2 = hhv[1][0] * kS; d = hhv[1][1] * kS;
    *(float2*)(hp + (1 * kQuads + q) * 2) = make_float2(a - d, b + c2);  // d45
    *(float2*)(hp + (4 * kQuads + q) * 2) = make_float2(a + d, b - c2);  // d135
    a = hlv[0][0] * kS; b = hlv[0][1] * kS; c2 = hlv[1][0] * kS; d = hlv[1][1] * kS;
    *(float2*)(hp + (2 * kQuads + q) * 2) = make_float2(a - d, b + c2);  // d75
    *(float2*)(hp + (3 * kQuads + q) * 2) = make_float2(a + d, b - c2);  // d105
  }
}

// ---------------------------------------------------------------------------
// WMMA: one wave computes C[32x32] = A[32x32(K)] * B[32(K)x32] with fp32
// V_WMMA_F32_16X16X4_F32 (2x2 C tiles, 8 K-steps of 4).
// Fragment layouts per CDNA5 ISA 7.12.2:
//   A 16x4 : lane<16 -> M=lane, VGPR v holds K = v + 2*(lane>>4)
//   B 4x16 : lane&15 -> N,      VGPR v holds K = v + 2*(lane>>4)
//   C 16x16: lane&15 -> N,      VGPR v holds M = v + 8*(lane>>4)
// ---------------------------------------------------------------------------
__device__ __forceinline__ v8f wmma4(v2f a, v2f b, v8f c) {
  return __builtin_amdgcn_wmma_f32_16x16x4_f32(false, a, false, b, (short)0, c,
                                               false, false);
}

__device__ __forceinline__ void gemm32_wave(const float* __restrict__ A,
                                            size_t am, size_t ak,
                                            const float* __restrict__ Bm,
                                            size_t bk, size_t bn,
                                            float* __restrict__ C, size_t cm,
                                            size_t cn) {
  const int lane = threadIdx.x & 31;
  const int half = lane >> 4;
  const int i16 = lane & 15;
  v8f c00 = {0, 0, 0, 0, 0, 0, 0, 0};
  v8f c01 = {0, 0, 0, 0, 0, 0, 0, 0};
  v8f c10 = {0, 0, 0, 0, 0, 0, 0, 0};
  v8f c11 = {0, 0, 0, 0, 0, 0, 0, 0};
#pragma unroll
  for (int kk = 0; kk < 8; ++kk) {
    const int k0 = kk * 4 + half * 2;
    v2f a0, a1, b0, b1;
    a0.x = A[(size_t)i16 * am + (size_t)k0 * ak];
    a0.y = A[(size_t)i16 * am + (size_t)(k0 + 1) * ak];
    a1.x = A[(size_t)(i16 + 16) * am + (size_t)k0 * ak];
    a1.y = A[(size_t)(i16 + 16) * am + (size_t)(k0 + 1) * ak];
    b0.x = Bm[(size_t)k0 * bk + (size_t)i16 * bn];
    b0.y = Bm[(size_t)(k0 + 1) * bk + (size_t)i16 * bn];
    b1.x = Bm[(size_t)k0 * bk + (size_t)(i16 + 16) * bn];
    b1.y = Bm[(size_t)(k0 + 1) * bk + (size_t)(i16 + 16) * bn];
    c00 = wmma4(a0, b0, c00);
    c01 = wmma4(a0, b1, c01);
    c10 = wmma4(a1, b0, c10);
    c11 = wmma4(a1, b1, c11);
  }
#pragma unroll
  for (int v = 0; v < 8; ++v) {
    const int m0 = v + 8 * half;
    C[(size_t)m0 * cm + (size_t)i16 * cn] = c00[v];
    C[(size_t)m0 * cm + (size_t)(i16 + 16) * cn] = c01[v];
    C[(size_t)(m0 + 16) * cm + (size_t)i16 * cn] = c10[v];
    C[(size_t)(m0 + 16) * cm + (size_t)(i16 + 16) * cn] = c11[v];
  }
}

// out_ll[b,o,x,y] = sum_i ll[b,i,x,y] * weight0[i,o,x,y]; one wave per pixel.
__global__ __launch_bounds__(256) void mix_ll(const float* __restrict__ ll,
                                              const float* __restrict__ w0,
                                              float* __restrict__ out_ll) {
  const int p = blockIdx.x * 8 + (threadIdx.x >> 5);  // pixel 0..4095
  gemm32_wave(ll + p, (size_t)kCi * kImg, (size_t)kImg,
              w0 + p, (size_t)kCo * kImg, (size_t)kImg,
              out_ll + p, (size_t)kCo * kImg, (size_t)kImg);
}

// out_h[b,c,o6,x,y,r] = sum_i highs[b,i,o6,x,y,r] * weight[1+2*o6+r][i,c,x,y]
// One wave per (o6, r, pixel) problem: 6*2*1024 = 12288 problems.
__global__ __launch_bounds__(256) void mix_h(const float* __restrict__ highs,
                                             const float* __restrict__ weight,
                                             float* __restrict__ out_h) {
  const int prob = blockIdx.x * 8 + (threadIdx.x >> 5);
  const int o6 = prob >> 11;
  const int rq = prob & 2047;
  const int r = rq >> 10;
  const int q = rq & 1023;
  const int xx = q >> 5, yy = q & 31;
  const size_t site = (size_t)o6 * (kQuads * 2) + (size_t)q * 2 + r;
  const float* A = highs + site;
  const float* Bm = weight + (size_t)(1 + 2 * o6 + r) * kCi * kCo * kImg +
                    (size_t)(xx * kW + yy);
  float* C = out_h + site;
  gemm32_wave(A, (size_t)kCi * kHiStride, (size_t)kHiStride,
              Bm, (size_t)kCo * kImg, (size_t)kImg,
              C, (size_t)kCo * kHiStride, (size_t)kHiStride);
}

// ---------------------------------------------------------------------------
// Kernel 3: inverse DTCWT. One workgroup per (b, co) output image.
// ---------------------------------------------------------------------------
__device__ __forceinline__ void build_band(float (*band)[kLds],
                                           const float* __restrict__ hp,
                                           int o1, int o2, int t) {
  for (int q = t; q < kQuads; q += 256) {
    const int i = q >> 5, j = q & 31;
    const float2 w1 = *(const float2*)(hp + (o1 * kQuads + q) * 2);
    const float2 w2 = *(const float2*)(hp + (o2 * kQuads + q) * 2);
    band[2 * i][2 * j] = (w1.x + w2.x) * kS;
    band[2 * i][2 * j + 1] = (w1.y + w2.y) * kS;
    band[2 * i + 1][2 * j] = (w1.y - w2.y) * kS;
    band[2 * i + 1][2 * j + 1] = (w2.x - w1.x) * kS;
  }
}

__device__ __forceinline__ void colf19_set(float (*dst)[kLds],
                                           const float (*src)[kLds],
                                           const float* g, int t) {
  for (int idx = t; idx < kImg; idx += 256) {
    const int r = idx >> 6, c = idx & 63;
    float a = 0.f;
#pragma unroll
    for (int k = 0; k < 19; ++k) a += g[k] * src[symi(r - 9 + k)][c];
    dst[r][c] = a;
  }
}

__device__ __forceinline__ void colf13_add(float (*dst)[kLds],
                                           const float (*src)[kLds],
                                           const float* g, int t) {
  for (int idx = t; idx < kImg; idx += 256) {
    const int r = idx >> 6, c = idx & 63;
    float a = dst[r][c];
#pragma unroll
    for (int k = 0; k < 13; ++k) a += g[k] * src[symi(r - 6 + k)][c];
    dst[r][c] = a;
  }
}

__global__ __launch_bounds__(256) void inv_dtcwt(const float* __restrict__ out_ll,
                                                 const float* __restrict__ out_h,
                                                 float* __restrict__ out) {
  __shared__ float stg[kH][kLds];
  __shared__ float lo[kH][kLds];
  __shared__ float hi[kH][kLds];
  const int image = blockIdx.x;  // b*kCo + co
  const int t = threadIdx.x;
  const float* llp = out_ll + (size_t)image * kImg;
  const float* hp = out_h + (size_t)image * kHiStride;

  // Stage ll with b128 loads.
  for (int idx = t * 4; idx < kImg; idx += 1024) {
    const float4 v4 = *(const float4*)(llp + idx);
    const int r = idx >> 6, c0 = idx & 63;
    stg[r][c0] = v4.x; stg[r][c0 + 1] = v4.y;
    stg[r][c0 + 2] = v4.z; stg[r][c0 + 3] = v4.w;
  }
  __syncthreads();
  colf19_set(lo, stg, cG0, t);  // lo = colf(ll, G0)
  __syncthreads();

  build_band(stg, hp, 0, 5, t);  // lh from (d15, d165)
  __syncthreads();
  colf13_add(lo, stg, cG1, t);  // lo += colf(lh, G1)
  __syncthreads();

  build_band(stg, hp, 2, 3, t);  // hl from (d75, d105)
  __syncthreads();
  colf19_set(hi, stg, cG0, t);  // hi = colf(hl, G0)
  __syncthreads();

  build_band(stg, hp, 1, 4, t);  // hh from (d45, d135)
  __syncthreads();
  colf13_add(hi, stg, cG1, t);  // hi += colf(hh, G1)
  __syncthreads();

  // out = rowf(lo, G0) + rowf(hi, G1); b128 stores.
  float* op = out + (size_t)image * kImg;
  for (int idx = t * 4; idx < kImg; idx += 1024) {
    const int r = idx >> 6, c0 = idx & 63;
    float acc[4] = {0.f, 0.f, 0.f, 0.f};
#pragma unroll
    for (int k = 0; k < 19; ++k) {
      const float w = cG0[k];
#pragma unroll
      for (int j = 0; j < 4; ++j) acc[j] += w * lo[r][symi(c0 + j - 9 + k)];
    }
#pragma unroll
    for (int k = 0; k < 13; ++k) {
      const float w = cG1[k];
#pragma unroll
      for (int j = 0; j < 4; ++j) acc[j] += w * hi[r][symi(c0 + j - 6 + k)];
    }
    float4 v4 = {acc[0], acc[1], acc[2], acc[3]};
    *(float4*)(op + idx) = v4;
  }
}

}  // namespace

extern "C" void kernel_launch(void* const* d_in, const int* in_sizes, int n_in,
                              void* d_out, int out_size, void* d_ws,
                              size_t ws_size, hipStream_t stream) {
  (void)in_sizes; (void)n_in; (void)out_size; (void)ws_size;
  const float* x = (const float*)d_in[0];
  const float* weight = (const float*)d_in[1];
  float* out = (float*)d_out;

  // Workspace partition (floats): ll | highs | out_ll | out_h  (128 MB total)
  float* ll = (float*)d_ws;
  float* highs = ll + (size_t)kB * kCi * kImg;            // 4,194,304
  float* out_ll = highs + (size_t)kB * kCi * kHiStride;   // +12,582,912
  float* out_h = out_ll + (size_t)kB * kCo * kImg;        // +4,194,304

  fwd_dtcwt<<<kB * kCi, 256, 0, stream>>>(x, ll, highs);
  mix_ll<<<kImg / 8, 256, 0, stream>>>(ll, weight, out_ll);               // 512
  mix_h<<<(6 * 2 * kQuads) / 8, 256, 0, stream>>>(highs, weight, out_h);  // 1536
  inv_dtcwt<<<kB * kCo, 256, 0, stream>>>(out_ll, out_h, out);
}